// TemporalCutoffModule_77103252898284
// MI455X (gfx1250) — compile-verified
//
#include <hip/hip_runtime.h>
#include <stdint.h>

// Problem dims (fixed by the reference)
#define B_DIM 512
#define N_DIM 64
#define S_DIM 2048
#define ROWS  16                 // b-rows per block (one WMMA M-tile)
#define XPAD  4                  // keeps 16B alignment, rotates banks by 4/row
#define XSTRIDE (S_DIM + XPAD)   // 2052 floats

typedef __attribute__((ext_vector_type(2))) float v2f;
typedef __attribute__((ext_vector_type(4))) float v4f;
typedef __attribute__((ext_vector_type(8))) float v8f;

// CDNA5 async copy: global -> LDS, 16 bytes per lane, tracked by ASYNCcnt.
__device__ __forceinline__ void async_b128(uint32_t lds_byte_addr, const void* gaddr) {
    asm volatile("global_load_async_to_lds_b128 %0, %1, off"
                 :: "v"(lds_byte_addr), "v"(gaddr)
                 : "memory");
}
__device__ __forceinline__ void wait_async0() {
    asm volatile("s_wait_asynccnt 0x0" ::: "memory");
}

extern "C" __global__ __launch_bounds__(256)
void temporal_cutoff_fused(const float* __restrict__ x,
                           const float* __restrict__ Wm,
                           const float* __restrict__ bias,
                           float* __restrict__ out) {
    extern __shared__ float smem[];
    float* sx = smem;                        // ROWS * XSTRIDE  (x tile)
    float* sw = smem + ROWS * XSTRIDE;       // S_DIM           (W row)
    float* sc = sw + S_DIM;                  // ROWS            (cutoffs)

    const int tid  = threadIdx.x;
    const int lane = tid & 31;
    // wave id is wave-uniform: force it into an SGPR so the WMMA K-loop has
    // scalar bounds (pure s_cmp/s_cbranch loop, EXEC stays all-1s as the
    // WMMA restriction requires — no lane-predicated loop machinery).
    const int wave = __builtin_amdgcn_readfirstlane(tid >> 5);

    const int tile = blockIdx.x;             // (B/ROWS)*N tiles
    const int n    = tile % N_DIM;
    const int b0   = (tile / N_DIM) * ROWS;

    // seed cutoffs with bias (same n for all 16 rows)
    if (tid < ROWS) sc[tid] = bias[n];

    // ---- async stage: W row + 16 x-rows into LDS ----
    {
        const float* wrow = Wm + (size_t)n * S_DIM;
        #pragma unroll
        for (int j = 0; j < 2; ++j) {
            int off = j * 1024 + tid * 4;
            async_b128((uint32_t)(uintptr_t)(sw + off), wrow + off);
        }
    }
    #pragma unroll
    for (int r = 0; r < ROWS; ++r) {
        const float* xrow = x + ((size_t)(b0 + r) * N_DIM + n) * S_DIM;
        #pragma unroll
        for (int j = 0; j < 2; ++j) {
            int off = j * 1024 + tid * 4;
            async_b128((uint32_t)(uintptr_t)(sx + r * XSTRIDE + off), xrow + off);
        }
    }
    wait_async0();
    __syncthreads();

    // ---- WMMA f32 16x16x4 dot products: cutoff[r] += x[r][:] . W[n][:] ----
    // A: 16x4 tile of x rows; B: W values broadcast across all 16 columns,
    // so every column of D holds the same partial dot per row.
    const int half  = lane >> 4;     // 0 = lanes 0-15 (K 0,1), 1 = lanes 16-31 (K 2,3)
    const int row16 = lane & 15;
    v8f acc = {};
    const int kbeg = wave * (S_DIM / 8);          // scalar
    const float* ap = sx + row16 * XSTRIDE + kbeg + half * 2;
    const float* bp = sw + kbeg + half * 2;
    #pragma unroll 8
    for (int k = 0; k < S_DIM / 8; k += 4) {
        v2f a  = *(const v2f*)(ap + k);
        v2f bm = *(const v2f*)(bp + k);
        acc = __builtin_amdgcn_wmma_f32_16x16x4_f32(false, a, false, bm,
                                                    (short)0, acc, false, false);
    }
    // Column 0 of C/D lives in lane 0 (rows 0-7) and lane 16 (rows 8-15).
    if ((lane & 15) == 0) {
        const int rbase = half * 8;
        #pragma unroll
        for (int j = 0; j < 8; ++j)
            atomicAdd(&sc[rbase + j], acc[j]);   // ds_add_f32, cross-wave combine
    }
    __syncthreads();

    // ---- gating pass: out = x * ((cutoff*(s-S)+1) > 0), coalesced stores ----
    const float lag0 = (float)(tid * 4 - S_DIM);  // one convert per thread
    #pragma unroll 4
    for (int r = 0; r < ROWS; ++r) {
        const float cut = sc[r];
        float* orow = out + ((size_t)(b0 + r) * N_DIM + n) * S_DIM;
        #pragma unroll
        for (int j = 0; j < 2; ++j) {
            int s = j * 1024 + tid * 4;
            v4f xv = *(const v4f*)(sx + r * XSTRIDE + s);
            v4f o;
            #pragma unroll
            for (int e = 0; e < 4; ++e) {
                float lag = lag0 + (float)(j * 1024 + e);    // negative lags
                float la  = fmaf(cut, lag, 1.0f);
                o[e] = (la > 0.0f) ? xv[e] : 0.0f;
            }
            *(v4f*)(orow + s) = o;
        }
    }
}

extern "C" void kernel_launch(void* const* d_in, const int* in_sizes, int n_in,
                              void* d_out, int out_size, void* d_ws, size_t ws_size,
                              hipStream_t stream) {
    const float* x  = (const float*)d_in[0];   // [B, N, S]
    const float* Wm = (const float*)d_in[1];   // [N, S]
    const float* b  = (const float*)d_in[2];   // [N]
    float* out = (float*)d_out;                // [B, N, S]

    const int blocks = (B_DIM / ROWS) * N_DIM; // 2048
    const size_t shmem = (size_t)(ROWS * XSTRIDE + S_DIM + ROWS) * sizeof(float); // ~139.6 KB
    hipLaunchKernelGGL(temporal_cutoff_fused, dim3(blocks), dim3(256), shmem, stream,
                       x, Wm, b, out);
}